// GAT_8787503087825
// MI455X (gfx1250) — compile-verified
//
#include <hip/hip_runtime.h>
#include <math.h>

typedef __attribute__((ext_vector_type(2))) float v2f;
typedef __attribute__((ext_vector_type(8))) float v8f;

#define GAT_NFEAT 512
#define GAT_NHID  64
#define GAT_HEADS 8
#define GAT_NCLASS 64

// ---------------------------------------------------------------------------
// Zero-fill (grid-stride). Accumulation buffers must be zero on every call.
// ---------------------------------------------------------------------------
__global__ void gat_zero_kernel(float* __restrict__ p, size_t n) {
  size_t i = (size_t)blockIdx.x * blockDim.x + threadIdx.x;
  size_t stride = (size_t)gridDim.x * blockDim.x;
  for (; i < n; i += stride) p[i] = 0.0f;
}

// ---------------------------------------------------------------------------
// FP32 WMMA GEMM: C[M x 64*gridDim.y] = A[M x 512] * B
//   B is head-blocked: block column-tile y uses B + y*512*64, ldb = 64.
//   (layer 2 launches with gridDim.y == 1, so the same kernel serves both.)
//   Each 128-thread block computes a 64x64 output tile; each wave32 owns a
//   16-row stripe = four v_wmma_f32_16x16x4_f32 accumulators, K swept 4/step.
// ---------------------------------------------------------------------------
__global__ __launch_bounds__(128) void gat_gemm_f32_wmma(
    const float* __restrict__ A, const float* __restrict__ B,
    float* __restrict__ C, int M, int ldc) {
  __shared__ float As[64][36];  // 64x32 tile, padded row stride (16B aligned)
  __shared__ float Bs[32][68];  // 32x64 tile, padded row stride (16B aligned)

  const int tid     = threadIdx.x;
  const int lane    = tid & 31;        // wave32
  const int wv      = tid >> 5;        // 0..3
  const int laneLow = lane & 15;
  const int kOff    = (lane >> 4) << 1;  // 0 for lanes 0-15, 2 for lanes 16-31
  const int rowBase = blockIdx.x * 64;
  const int colBase = blockIdx.y * 64;
  const float* Bblk = B + (size_t)blockIdx.y * (GAT_NFEAT * 64);

  v8f acc[4];
  acc[0] = {}; acc[1] = {}; acc[2] = {}; acc[3] = {};

  const int aCol = (tid & 7)  * 4;   // 0..28
  const int bCol = (tid & 15) * 4;   // 0..60

  for (int k0 = 0; k0 < GAT_NFEAT; k0 += 32) {
    // ---- stage A tile (64x32) : 4 float4 per thread, row-clamped ----
#pragma unroll
    for (int p = 0; p < 4; ++p) {
      int row = (tid >> 3) + p * 16;
      int gRow = rowBase + row;
      if (gRow >= M) gRow = M - 1;  // clamp: keeps EXEC full for WMMA
      *(float4*)&As[row][aCol] =
          *(const float4*)(A + (size_t)gRow * GAT_NFEAT + k0 + aCol);
    }
    // ---- stage B tile (32x64) ----
#pragma unroll
    for (int p = 0; p < 4; ++p) {
      int row = (tid >> 4) + p * 8;
      *(float4*)&Bs[row][bCol] =
          *(const float4*)(Bblk + (size_t)(k0 + row) * 64 + bCol);
    }
    // prefetch next A tile while we compute this one (global_prefetch_b8)
    if (k0 + 32 < GAT_NFEAT) {
      int gRow = rowBase + (tid >> 3);
      if (gRow >= M) gRow = M - 1;
      __builtin_prefetch(A + (size_t)gRow * GAT_NFEAT + k0 + 32 + aCol, 0, 3);
    }
    __syncthreads();

    // ---- 8 K-steps of 4, four 16x16 col-tiles per wave ----
#pragma unroll
    for (int kk = 0; kk < 32; kk += 4) {
      v2f a;
      a.x = As[wv * 16 + laneLow][kk + kOff];
      a.y = As[wv * 16 + laneLow][kk + kOff + 1];
#pragma unroll
      for (int n = 0; n < 4; ++n) {
        v2f b;
        b.x = Bs[kk + kOff][n * 16 + laneLow];
        b.y = Bs[kk + kOff + 1][n * 16 + laneLow];
        acc[n] = __builtin_amdgcn_wmma_f32_16x16x4_f32(
            false, a, false, b, (short)0, acc[n], false, false);
      }
    }
    __syncthreads();
  }

  // ---- write back: VGPR r -> rows (M=r | M=r+8), N = laneLow ----
#pragma unroll
  for (int n = 0; n < 4; ++n) {
#pragma unroll
    for (int r = 0; r < 8; ++r) {
      int row = rowBase + wv * 16 + r + ((lane >> 4) << 3);
      if (row < M)
        C[(size_t)row * ldc + colBase + n * 16 + laneLow] = acc[n][r];
    }
  }
}

// ---------------------------------------------------------------------------
// alpha[i][h] = H[i, h*64:+64] . a[h][0:64] ; beta uses a[h][64:128]
// ---------------------------------------------------------------------------
__global__ void gat_alpha_beta_kernel(
    const float* __restrict__ H, const float* __restrict__ a,
    float* __restrict__ alpha, float* __restrict__ beta, int n, int heads) {
  int t = blockIdx.x * blockDim.x + threadIdx.x;
  if (t >= n * heads) return;
  int node = t / heads;
  int h = t - node * heads;
  const float* hv = H + (size_t)node * heads * 64 + h * 64;
  const float* av = a + h * 128;
  float sa = 0.0f, sb = 0.0f;
#pragma unroll
  for (int d = 0; d < 64; d += 4) {
    float4 hh = *(const float4*)(hv + d);
    float4 a1 = *(const float4*)(av + d);
    float4 a2 = *(const float4*)(av + 64 + d);
    sa += hh.x * a1.x + hh.y * a1.y + hh.z * a1.z + hh.w * a1.w;
    sb += hh.x * a2.x + hh.y * a2.y + hh.z * a2.z + hh.w * a2.w;
  }
  alpha[t] = sa;
  beta[t] = sb;
}

// ---------------------------------------------------------------------------
// Edge scatter: one wave32 per edge. Lanes < HEADS produce the attention
// weight and bump rowsum; then HEADS*2 coalesced chunks of 32 lanes each do
// atomicAdd(agg[row][idx], w[head] * H[col][idx]).
// ---------------------------------------------------------------------------
template <int HEADS>
__global__ __launch_bounds__(256) void gat_edge_kernel(
    const int* __restrict__ rows, const int* __restrict__ cols,
    const float* __restrict__ alpha, const float* __restrict__ beta,
    const float* __restrict__ H, float* __restrict__ rowsum,
    float* __restrict__ agg, int E) {
  int wave = blockIdx.x * (blockDim.x >> 5) + (threadIdx.x >> 5);
  int lane = threadIdx.x & 31;
  if (wave >= E) return;
  const int r = rows[wave];
  const int c = cols[wave];

  float w[HEADS];
#pragma unroll
  for (int h = 0; h < HEADS; ++h) {
    float s = alpha[r * HEADS + h] + beta[c * HEADS + h];
    s = s > 0.0f ? s : 0.2f * s;   // LeakyReLU(0.2)
    w[h] = __expf(-s);             // edge weight
  }
  if (lane < HEADS) {
    float s = alpha[r * HEADS + lane] + beta[c * HEADS + lane];
    s = s > 0.0f ? s : 0.2f * s;
    atomicAdd(&rowsum[r * HEADS + lane], __expf(-s));
  }

  const float* hc = H + (size_t)c * (HEADS * 64);
  float* ar = agg + (size_t)r * (HEADS * 64);
#pragma unroll
  for (int ch = 0; ch < HEADS * 2; ++ch) {  // 64 dims/head, 32 lanes/chunk
    int idx = ch * 32 + lane;
    atomicAdd(&ar[idx], w[ch >> 1] * hc[idx]);
  }
}

// ---------------------------------------------------------------------------
// xc = ELU(agg / rowsum)  in place (layer 1)
// ---------------------------------------------------------------------------
__global__ void gat_norm_elu_kernel(float* __restrict__ agg,
                                    const float* __restrict__ rowsum, int n) {
  int t = blockIdx.x * blockDim.x + threadIdx.x;
  if (t >= n * GAT_HEADS * 64) return;
  int node = t >> 9;          // / 512
  int h = (t >> 6) & 7;       // head
  float v = agg[t] / rowsum[node * GAT_HEADS + h];
  agg[t] = v > 0.0f ? v : (__expf(v) - 1.0f);
}

// out = out / rowsum (layer 2, no activation)
__global__ void gat_norm_kernel(float* __restrict__ out,
                                const float* __restrict__ rowsum, int n) {
  int t = blockIdx.x * blockDim.x + threadIdx.x;
  if (t >= n * GAT_NCLASS) return;
  out[t] = out[t] / rowsum[t >> 6];
}

// ---------------------------------------------------------------------------
extern "C" void kernel_launch(void* const* d_in, const int* in_sizes, int n_in,
                              void* d_out, int out_size, void* d_ws,
                              size_t ws_size, hipStream_t stream) {
  const float* x       = (const float*)d_in[0];  // [N, 512]
  const float* Ws      = (const float*)d_in[1];  // [8, 512, 64]
  const float* a_heads = (const float*)d_in[2];  // [8, 128]
  const float* W_out   = (const float*)d_in[3];  // [512, 64]
  const float* a_out   = (const float*)d_in[4];  // [128]
  const int*   eidx    = (const int*)d_in[5];    // [2, E]
  float* out = (float*)d_out;                    // [N, 64]

  const int N = in_sizes[0] / GAT_NFEAT;
  const int E = in_sizes[5] / 2;
  const int* e_row = eidx;
  const int* e_col = eidx + E;

  // workspace layout (floats); rowsum1+agg1 kept adjacent for one zero pass
  float* ws = (float*)d_ws;
  size_t off = 0;
  float* H       = ws + off; off += (size_t)N * 512;
  float* alpha1  = ws + off; off += (size_t)N * 8;
  float* beta1   = ws + off; off += (size_t)N * 8;
  float* rowsum1 = ws + off; off += (size_t)N * 8;
  float* agg1    = ws + off; off += (size_t)N * 512;  // becomes xc in place
  float* H2      = ws + off; off += (size_t)N * 64;
  float* alpha2  = ws + off; off += (size_t)N;
  float* beta2   = ws + off; off += (size_t)N;
  float* rowsum2 = ws + off; off += (size_t)N;
  (void)ws_size; (void)n_in; (void)out_size;

  // 0) zero the accumulators (rowsum1|agg1 contiguous, rowsum2, d_out)
  gat_zero_kernel<<<2048, 256, 0, stream>>>(rowsum1, (size_t)N * (8 + 512));
  gat_zero_kernel<<<256, 256, 0, stream>>>(rowsum2, (size_t)N);
  gat_zero_kernel<<<2048, 256, 0, stream>>>(out, (size_t)N * GAT_NCLASS);

  const int mblocks = (N + 63) / 64;

  // 1) H[N,512] = x @ Ws   (8 head column-tiles)
  gat_gemm_f32_wmma<<<dim3(mblocks, GAT_HEADS), 128, 0, stream>>>(
      x, Ws, H, N, 512);

  // 2) per-node attention halves
  gat_alpha_beta_kernel<<<(N * GAT_HEADS + 255) / 256, 256, 0, stream>>>(
      H, a_heads, alpha1, beta1, N, GAT_HEADS);

  // 3) edge scatter, layer 1 (8 waves/block -> 8 edges/block)
  gat_edge_kernel<GAT_HEADS><<<(E + 7) / 8, 256, 0, stream>>>(
      e_row, e_col, alpha1, beta1, H, rowsum1, agg1, E);

  // 4) xc = ELU(agg1 / rowsum1), in place
  gat_norm_elu_kernel<<<(N * 512 + 255) / 256, 256, 0, stream>>>(
      agg1, rowsum1, N);

  // 5) H2[N,64] = xc @ W_out
  gat_gemm_f32_wmma<<<dim3(mblocks, 1), 128, 0, stream>>>(
      agg1, W_out, H2, N, 64);

  // 6) per-node attention halves, layer 2
  gat_alpha_beta_kernel<<<(N + 255) / 256, 256, 0, stream>>>(
      H2, a_out, alpha2, beta2, N, 1);

  // 7) edge scatter, layer 2 (accumulate straight into d_out)
  gat_edge_kernel<1><<<(E + 7) / 8, 256, 0, stream>>>(
      e_row, e_col, alpha2, beta2, H2, rowsum2, out, E);

  // 8) final normalization
  gat_norm_kernel<<<(N * GAT_NCLASS + 255) / 256, 256, 0, stream>>>(
      out, rowsum2, N);
}